// Sentence_Rep_42855183679779
// MI455X (gfx1250) — compile-verified
//
#include <hip/hip_runtime.h>
#include <hip/hip_bf16.h>

// mean over seq axis of [4096, 128, 512] f32 -> [4096, 512] f32
// Implemented as D = A x B + C with A = uniform (1/128), using
// V_WMMA_F32_16X16X4_F32 (CDNA5, wave32). Each wave handles one
// (batch, 16-column embedding tile); 32 WMMAs (4 chains of 8) cover seq=128.

typedef __attribute__((ext_vector_type(2))) float v2f;
typedef __attribute__((ext_vector_type(8))) float v8f;

#define BATCH   4096
#define SEQLEN  128
#define EDIM    512
#define ETILES  (EDIM / 16)          // 32 e-tiles per batch
#define WAVES_PER_BLOCK 8            // 256 threads = 8 wave32

__global__ __launch_bounds__(256)
void Sentence_Rep_mean_wmma(const float* __restrict__ x,
                            float* __restrict__ out) {
    const int wave = blockIdx.x * WAVES_PER_BLOCK + (threadIdx.x >> 5);
    const int lane = threadIdx.x & 31;

    const int b  = wave >> 5;        // batch index   [0, 4096)
    const int et = wave & 31;        // e-tile index  [0, 32)

    const int n  = lane & 15;        // output column within tile (N = lane%16)
    const int kh = lane >> 4;        // which half of the K slots this lane fills

    // Per-lane base: row (s = 2*kh) of this batch, column e0+n.
    // All 64 loads below are immediate offsets from this base.
    const float* base = x + ((size_t)b * SEQLEN + 2 * kh) * EDIM + et * 16 + n;

    // Uniform A = 1/128 in every slot -> D rows all equal the column means.
    v2f a;
    a.x = 1.0f / 128.0f;
    a.y = 1.0f / 128.0f;

    v8f acc0 = {}; v8f acc1 = {}; v8f acc2 = {}; v8f acc3 = {};

    // 32 WMMA steps, K=4 each: step j covers seq rows [4j, 4j+4).
    // B register pair per lane: v0 = X[s0 + 2*kh], v1 = X[s0 + 2*kh + 1].
    // (K-permutation inside B is irrelevant because A is uniform.)
#pragma unroll
    for (int j = 0; j < 32; j += 4) {
        v2f b0, b1, b2, b3;
        b0.x = base[(j + 0) * 4 * EDIM];
        b0.y = base[(j + 0) * 4 * EDIM + EDIM];
        b1.x = base[(j + 1) * 4 * EDIM];
        b1.y = base[(j + 1) * 4 * EDIM + EDIM];
        b2.x = base[(j + 2) * 4 * EDIM];
        b2.y = base[(j + 2) * 4 * EDIM + EDIM];
        b3.x = base[(j + 3) * 4 * EDIM];
        b3.y = base[(j + 3) * 4 * EDIM + EDIM];

        acc0 = __builtin_amdgcn_wmma_f32_16x16x4_f32(
            false, a, false, b0, (short)0, acc0, false, false);
        acc1 = __builtin_amdgcn_wmma_f32_16x16x4_f32(
            false, a, false, b1, (short)0, acc1, false, false);
        acc2 = __builtin_amdgcn_wmma_f32_16x16x4_f32(
            false, a, false, b2, (short)0, acc2, false, false);
        acc3 = __builtin_amdgcn_wmma_f32_16x16x4_f32(
            false, a, false, b3, (short)0, acc3, false, false);
    }

    // Combine the 4 independent accumulator chains. Every row of D is
    // identical (uniform A), so component 0 of any lane holds the mean for
    // column N = lane%16.
    v8f s = acc0 + acc1 + acc2 + acc3;

    if (lane < 16) {
        out[(size_t)b * EDIM + et * 16 + lane] = s[0];
    }
}

extern "C" void kernel_launch(void* const* d_in, const int* in_sizes, int n_in,
                              void* d_out, int out_size, void* d_ws, size_t ws_size,
                              hipStream_t stream) {
    (void)in_sizes; (void)n_in; (void)out_size; (void)d_ws; (void)ws_size;
    const float* x = (const float*)d_in[0];
    float* out = (float*)d_out;

    // One wave per (batch, e-tile): 4096 * 32 = 131072 waves.
    const int total_waves = BATCH * ETILES;
    const int blocks = total_waves / WAVES_PER_BLOCK;   // 16384
    Sentence_Rep_mean_wmma<<<blocks, 256, 0, stream>>>(x, out);
}